// SmartLinearAppearance_83476984365256
// MI455X (gfx1250) — compile-verified
//
#include <hip/hip_runtime.h>
#include <hip/hip_bf16.h>

// ---------------------------------------------------------------------------
// Problem constants (from the reference):
//   embs [B=8, N=256, T=32, P=7, E=256] f32
//   vis  [B,N,T,P] f32,  masks [B,N,T] bool(1B)
//   W [1024, 1799] f32,  b [1024] f32
//   out [B,N,1024] f32,  rows = B*N = 2048, feat_dim = 7*256+7 = 1799
// ---------------------------------------------------------------------------
#define ROWS     2048       // B*N
#define TFRAMES  32
#define PP       7
#define EE       256
#define FEATDIM  1799
#define KP       1824       // feat_dim padded to multiple of 32 (zero-filled)
#define TOKDIM   1024
#define ALPHA_C  0.9f

typedef __attribute__((ext_vector_type(2))) float v2f;
typedef __attribute__((ext_vector_type(8))) float v8f;
typedef __attribute__((ext_vector_type(4))) int   v4i;

typedef __attribute__((address_space(1))) v4i g_v4i;   // global
typedef __attribute__((address_space(3))) v4i l_v4i;   // LDS

// ---------------------------------------------------------------------------
// CDNA5 async global->LDS copy (ASYNCcnt-tracked, no VGPR round trip).
// Builtin signature (per clang diag): (v4i AS1* src, v4i AS3* dst, Ii, Ii).
// ---------------------------------------------------------------------------
__device__ __forceinline__ void async_copy_b128(const float* __restrict__ gsrc,
                                                float* lds_dst) {
#if __has_builtin(__builtin_amdgcn_global_load_async_to_lds_b128)
    __builtin_amdgcn_global_load_async_to_lds_b128(
        (g_v4i*)gsrc, (l_v4i*)lds_dst, /*offset=*/0, /*cpol=*/0);
#else
    asm volatile("global_load_async_to_lds_b128 %0, %1, off"
                 :: "v"((unsigned)(unsigned long long)lds_dst), "v"(gsrc)
                 : "memory");
#endif
}

__device__ __forceinline__ void wait_asynccnt_le6() {
#if __has_builtin(__builtin_amdgcn_s_wait_asynccnt)
    __builtin_amdgcn_s_wait_asynccnt(6);
#else
    asm volatile("s_wait_asynccnt 0x6" ::: "memory");
#endif
}
__device__ __forceinline__ void wait_asynccnt_zero() {
#if __has_builtin(__builtin_amdgcn_s_wait_asynccnt)
    __builtin_amdgcn_s_wait_asynccnt(0);
#else
    asm volatile("s_wait_asynccnt 0x0" ::: "memory");
#endif
}

// ---------------------------------------------------------------------------
// Kernel 1: temporal EMA scan ("smart"). One block per (b,n) row; lane = e.
// State lives in registers; embs streamed once (~470 MB -> ~20 us at 23.3
// TB/s), fully coalesced over E. Writes padded feature row [KP] and any-mask.
// ---------------------------------------------------------------------------
__global__ __launch_bounds__(256)
void smart_fuse_kernel(const float* __restrict__ embs,
                       const float* __restrict__ vis,
                       const unsigned char* __restrict__ masks,
                       float* __restrict__ feats,      // [ROWS, KP]
                       int* __restrict__ flags)        // [ROWS]
{
    const int bn = blockIdx.x;        // 0..2047
    const int e  = threadIdx.x;       // 0..255

    float ce[PP], cv[PP];
#pragma unroll
    for (int p = 0; p < PP; ++p) { ce[p] = 0.0f; cv[p] = 0.0f; }

    const float* embBase = embs + (size_t)bn * TFRAMES * PP * EE;
    const float* visBase = vis  + (size_t)bn * TFRAMES * PP;
    const unsigned char* mBase = masks + (size_t)bn * TFRAMES;

    int any = 0;
    for (int t = 0; t < TFRAMES; ++t) {
        const bool m = (mBase[t] != 0);
        any |= (int)m;
#pragma unroll
        for (int p = 0; p < PP; ++p) {
            const float dv = visBase[t * PP + p];                 // broadcast load
            const float de = embBase[(size_t)(t * PP + p) * EE + e];
            const float cvp  = cv[p];
            const float x    = ((cvp != 0.0f) != (dv != 0.0f)) ? 1.0f : 0.0f;
            const float prod = cvp * dv;
            const float ema_t = prod * ALPHA_C + x * cvp;
            const float ema_d = prod * (1.0f - ALPHA_C) + x * dv;
            const float se = ema_t * ce[p] + ema_d * de;
            const float sv = fmaxf(cvp, dv);
            if (m) { ce[p] = se; cv[p] = sv; }
        }
    }

    float* row = feats + (size_t)bn * KP;
#pragma unroll
    for (int p = 0; p < PP; ++p) row[p * EE + e] = ce[p];
    if (e < 32) row[PP * EE + e] = (e < PP) ? cv[e] : 0.0f;   // vis + zero pad
    if (e == 0) flags[bn] = any;
}

// ---------------------------------------------------------------------------
// Kernel 2: pad-copy W [1024,1799] -> Wp [1024,KP] (zero pad K tail) so the
// GEMM gets aligned b128 traffic and no K-edge logic.
// ---------------------------------------------------------------------------
__global__ __launch_bounds__(256)
void pack_w_kernel(const float* __restrict__ W, float* __restrict__ Wp)
{
    const int idx = blockIdx.x * 256 + threadIdx.x;
    const int total = TOKDIM * KP;
    if (idx >= total) return;
    const int r = idx / KP;
    const int c = idx - r * KP;
    Wp[idx] = (c < FEATDIM) ? W[(size_t)r * FEATDIM + c] : 0.0f;
}

// ---------------------------------------------------------------------------
// Kernel 3: fp32 WMMA GEMM.  out[m,n] = sum_k feats[m,k] * Wp[n,k] + b[n],
// zeroed where !flags[m].  Both operands K-major (NT gemm).
// Workgroup = 256 threads = 8 wave32, tile 128(M) x 64(N).  K is staged in
// 32-float chunks via DOUBLE-BUFFERED async global->LDS copies: the next
// chunk's 6 b128 async loads per thread are issued before computing on the
// current chunk; s_wait_asynccnt<=6 drains only the current chunk (async
// loads complete in order).  Waves in a 4x2 grid, each computing 32x32 via
// 2x2 v_wmma_f32_16x16x4_f32 accumulators fed by ds_load of LDS fragments.
// ---------------------------------------------------------------------------
#define TM 128
#define TN 64
#define KB 32
#define LDA (KB + 4)   // +4 floats: keeps 16B alignment, breaks bank stride
#define NCHUNK (KP / KB)   // 57

static __device__ inline v8f wmma_f32(v2f a, v2f b, v8f c) {
    return __builtin_amdgcn_wmma_f32_16x16x4_f32(
        /*neg_a=*/false, a, /*neg_b=*/false, b,
        /*c_mod=*/(short)0, c, /*reuse_a=*/false, /*reuse_b=*/false);
}

__global__ __launch_bounds__(256)
void wmma_gemm_kernel(const float* __restrict__ feats,   // [ROWS, KP]
                      const float* __restrict__ Wp,      // [TOKDIM, KP]
                      const float* __restrict__ bias,    // [TOKDIM]
                      const int* __restrict__ flags,     // [ROWS]
                      float* __restrict__ out)           // [ROWS, TOKDIM]
{
    __shared__ float Als[2][TM * LDA];
    __shared__ float Bls[2][TN * LDA];

    const int tid    = threadIdx.x;
    const int lane   = tid & 31;
    const int waveId = tid >> 5;          // 0..7
    const int wm     = waveId & 3;        // 0..3 -> M offset wm*32
    const int wn     = waveId >> 2;       // 0..1 -> N offset wn*32
    const int hlf    = lane >> 4;         // K-pair select per ISA f32 A layout
    const int l15    = lane & 15;

    const int mTile = blockIdx.x * TM;    // gridDim.x = ROWS/TM = 16
    const int nTile = blockIdx.y * TN;    // gridDim.y = TOKDIM/TN = 16

    // per-thread staging coordinates (same pattern every chunk)
    const int ra0 = (tid + 0 * 256) >> 3, ca0 = (tid + 0 * 256) & 7;
    const int ra1 = (tid + 1 * 256) >> 3, ca1 = (tid + 1 * 256) & 7;
    const int ra2 = (tid + 2 * 256) >> 3, ca2 = (tid + 2 * 256) & 7;
    const int ra3 = (tid + 3 * 256) >> 3, ca3 = (tid + 3 * 256) & 7;

    auto issue_chunk = [&](int k0, int buf) {
        // A tile: 128 rows x 32 floats = 1024 b128 -> 4 per thread
        async_copy_b128(feats + (size_t)(mTile + ra0) * KP + k0 + ca0 * 4,
                        &Als[buf][ra0 * LDA + ca0 * 4]);
        async_copy_b128(feats + (size_t)(mTile + ra1) * KP + k0 + ca1 * 4,
                        &Als[buf][ra1 * LDA + ca1 * 4]);
        async_copy_b128(feats + (size_t)(mTile + ra2) * KP + k0 + ca2 * 4,
                        &Als[buf][ra2 * LDA + ca2 * 4]);
        async_copy_b128(feats + (size_t)(mTile + ra3) * KP + k0 + ca3 * 4,
                        &Als[buf][ra3 * LDA + ca3 * 4]);
        // B tile: 64 rows x 32 floats = 512 b128 -> 2 per thread
        async_copy_b128(Wp + (size_t)(nTile + ra0) * KP + k0 + ca0 * 4,
                        &Bls[buf][ra0 * LDA + ca0 * 4]);
        async_copy_b128(Wp + (size_t)(nTile + ra1) * KP + k0 + ca1 * 4,
                        &Bls[buf][ra1 * LDA + ca1 * 4]);
    };

    v8f acc[2][2];
#pragma unroll
    for (int i = 0; i < 2; ++i)
#pragma unroll
        for (int j = 0; j < 2; ++j) acc[i][j] = (v8f){0,0,0,0,0,0,0,0};

    issue_chunk(0, 0);                    // prologue: chunk 0 in flight

    for (int c = 0; c < NCHUNK; ++c) {
        const int buf = c & 1;
        const bool more = (c + 1 < NCHUNK);
        if (more) issue_chunk((c + 1) * KB, buf ^ 1);   // overlap next copy

        // drain current chunk's 6 async loads (in-order completion), keep
        // the next chunk's 6 in flight; then workgroup barrier.
        if (more) wait_asynccnt_le6(); else wait_asynccnt_zero();
        __syncthreads();

        const float* __restrict__ As = Als[buf];
        const float* __restrict__ Bs = Bls[buf];
#pragma unroll
        for (int kk = 0; kk < KB; kk += 4) {
            const int k = kk + hlf * 2;   // lanes 0-15: K=0,1 ; lanes 16-31: K=2,3
            const int ar0 = (wm * 32 + l15) * LDA + k;
            const int ar1 = ar0 + 16 * LDA;
            const int br0 = (wn * 32 + l15) * LDA + k;
            const int br1 = br0 + 16 * LDA;
            v2f a0 = { As[ar0], As[ar0 + 1] };
            v2f a1 = { As[ar1], As[ar1 + 1] };
            v2f b0 = { Bs[br0], Bs[br0 + 1] };
            v2f b1 = { Bs[br1], Bs[br1 + 1] };
            acc[0][0] = wmma_f32(a0, b0, acc[0][0]);
            acc[0][1] = wmma_f32(a0, b1, acc[0][1]);
            acc[1][0] = wmma_f32(a1, b0, acc[1][0]);
            acc[1][1] = wmma_f32(a1, b1, acc[1][1]);
        }
        __syncthreads();   // all waves done reading buf before it is re-filled
    }

    // --- epilogue: bias add + row mask, per ISA 16x16 f32 C/D layout:
    //     VGPR r, lane half h -> M = r + h*8 ; N = lane&15
#pragma unroll
    for (int j = 0; j < 2; ++j) {
        const int ncol = nTile + wn * 32 + j * 16 + l15;
        const float bv = bias[ncol];
#pragma unroll
        for (int i = 0; i < 2; ++i) {
            const int mrowBase = mTile + wm * 32 + i * 16 + hlf * 8;
#pragma unroll
            for (int r = 0; r < 8; ++r) {
                const int mrow = mrowBase + r;
                const float v = flags[mrow] ? (acc[i][j][r] + bv) : 0.0f;
                out[(size_t)mrow * TOKDIM + ncol] = v;
            }
        }
    }
}

// ---------------------------------------------------------------------------
// Host-side launcher. Workspace layout (all fp32 unless noted):
//   [0)                 feats  ROWS*KP           = 14,942,208 B
//   [featsB)            Wp     TOKDIM*KP         =  7,471,104 B
//   [featsB+WpB)        flags  ROWS int          =      8,192 B
// ---------------------------------------------------------------------------
extern "C" void kernel_launch(void* const* d_in, const int* in_sizes, int n_in,
                              void* d_out, int out_size, void* d_ws, size_t ws_size,
                              hipStream_t stream) {
    const float*         embs  = (const float*)d_in[0];
    const float*         vis   = (const float*)d_in[1];
    const unsigned char* masks = (const unsigned char*)d_in[2];
    const float*         W     = (const float*)d_in[3];
    const float*         bias  = (const float*)d_in[4];
    float*               out   = (float*)d_out;

    char* ws = (char*)d_ws;
    const size_t featsB = (size_t)ROWS * KP * sizeof(float);
    const size_t WpB    = (size_t)TOKDIM * KP * sizeof(float);
    float* feats = (float*)(ws);
    float* Wp    = (float*)(ws + featsB);
    int*   flags = (int*)(ws + featsB + WpB);

    // 1) temporal EMA scan -> padded feature rows + any-mask flags
    smart_fuse_kernel<<<ROWS, 256, 0, stream>>>(embs, vis, masks, feats, flags);

    // 2) pad W into [TOKDIM, KP]
    const int wTotal = TOKDIM * KP;
    pack_w_kernel<<<(wTotal + 255) / 256, 256, 0, stream>>>(W, Wp);

    // 3) fp32 WMMA GEMM + bias + mask (double-buffered async LDS staging)
    dim3 grid(ROWS / TM, TOKDIM / TN);
    wmma_gemm_kernel<<<grid, 256, 0, stream>>>(feats, Wp, bias, flags, out);
}